// GCN_43044162241229
// MI455X (gfx1250) — compile-verified
//
#include <hip/hip_runtime.h>

typedef __attribute__((ext_vector_type(2))) float v2f;
typedef __attribute__((ext_vector_type(8))) float v8f;

#define NFEAT  256
#define NHID   128
#define NCLASS 8
#define MROWS  5          // 16-row tiles per wave -> 80 rows (50000 = 625 * 80)

#if defined(__has_builtin)
#if __has_builtin(__builtin_amdgcn_global_load_async_to_lds_b32) && \
    __has_builtin(__builtin_amdgcn_s_wait_asynccnt)
#define HAVE_ASYNC_LDS 1
#endif
#endif

// ---------------- small utility kernels ----------------

__global__ void zero_f32(float* __restrict__ p, int n) {
    int i = blockIdx.x * blockDim.x + threadIdx.x;
    if (i < n) p[i] = 0.0f;
}

__global__ void deg_accum(const int* __restrict__ dst, float* __restrict__ deg, int E) {
    int e = blockIdx.x * blockDim.x + threadIdx.x;
    if (e < E) atomicAdd(&deg[dst[e]], 1.0f);
}

// deg includes +1 self loop; deg >= 1 so rsqrt(max(deg,1)) == rsqrt(deg)
__global__ void deg_inv_sqrt_k(float* __restrict__ d, int N) {
    int i = blockIdx.x * blockDim.x + threadIdx.x;
    if (i < N) d[i] = rsqrtf(d[i] + 1.0f);
}

// ---------------- fp32 WMMA GEMM ----------------
// H[M x ldH] = X[M x K] @ W[K x Wn] (+ optional bias).
// One wave per block; wave computes an 80-row x 16-col output slab
// (5 WMMA accumulators sharing each B fragment).
// grid.x = M/80 (M must be a multiple of 80), grid.y = ceil(Wn/16).
// B tile is staged into LDS (async copy on CDNA5), pre-masked for Wn<16 tiles.
__global__ void __launch_bounds__(32) gemm_wmma_f32(
    const float* __restrict__ X, const float* __restrict__ W,
    float* __restrict__ H, int K, int Wn, int ldH,
    const float* __restrict__ bias)
{
    __shared__ float bs[NFEAT * 16];   // max K = 256 -> 16 KB

    const int lane = threadIdx.x & 31;
    const int l    = lane & 15;        // position within 16-lane half
    const int g    = lane >> 4;        // lane group: K-pair selector

    const int cb0 = blockIdx.y * 16;
    const int cb  = cb0 + l;           // B/C column this lane supplies
    const bool cvalid = (cb < Wn);

    const int elems = K * 16;          // multiple of 32 -> no staging divergence

    if (cb0 + 16 <= Wn) {              // full tile: fast path
#ifdef HAVE_ASYNC_LDS
        for (int i = lane; i < elems; i += 32) {
            const int k = i >> 4;
            const int c = i & 15;
            __builtin_amdgcn_global_load_async_to_lds_b32(
                (__attribute__((address_space(1))) int*)(W + (size_t)k * Wn + cb0 + c),
                (__attribute__((address_space(3))) int*)&bs[i],
                0, 0);
        }
        __builtin_amdgcn_s_wait_asynccnt(0);
#else
        for (int i = lane; i < elems; i += 32) {
            const int k = i >> 4;
            const int c = i & 15;
            bs[i] = W[(size_t)k * Wn + cb0 + c];
        }
#endif
    } else {                           // ragged tile (fc, Wn=8): mask with zeros
        for (int i = lane; i < elems; i += 32) {
            const int k  = i >> 4;
            const int c  = i & 15;
            const int cc = cb0 + c;
            bs[i] = (cc < Wn) ? W[(size_t)k * Wn + cc] : 0.0f;
        }
    }
    __syncthreads();                   // single-wave WG: barrier -> S_NOP, enforces DS wait

    const int rowbase = blockIdx.x * (16 * MROWS);
    const int row     = rowbase + l;   // A row for tile t is row + 16*t

    v8f acc[MROWS] = {};
    for (int k = 0; k < K; k += 4) {
        const int ka = k + 2 * g;      // this lane-group's K pair
        v2f b;
        b.x = bs[ka * 16 + l];
        b.y = bs[ka * 16 + 16 + l];
#pragma unroll
        for (int t = 0; t < MROWS; ++t) {
            const float* __restrict__ ap = X + (size_t)(row + 16 * t) * K + ka;
            v2f a;
            a.x = ap[0];
            a.y = ap[1];
            acc[t] = __builtin_amdgcn_wmma_f32_16x16x4_f32(
                /*neg_a=*/false, a, /*neg_b=*/false, b,
                /*c_mod=*/(short)0, acc[t], /*reuse_a=*/false, /*reuse_b=*/false);
        }
    }

    if (cvalid) {
        const float bv = bias ? bias[cb] : 0.0f;
        float* __restrict__ hcol = H + cb;
#pragma unroll
        for (int t = 0; t < MROWS; ++t) {
            const int rb = rowbase + 16 * t + 8 * g;   // C layout: VGPR v -> row rb+v
#pragma unroll
            for (int v = 0; v < 8; ++v) {
                hcol[(size_t)(rb + v) * ldH] = acc[t][v] + bv;
            }
        }
    }
}

// ---------------- aggregation ----------------

// out[i][f] = h[i][f] * d[i]^2   (self-loop term; also initializes the buffer)
__global__ void agg_init(const float* __restrict__ h, const float* __restrict__ dinv,
                         float* __restrict__ out, int total, int F) {
    int idx = blockIdx.x * blockDim.x + threadIdx.x;
    if (idx < total) {
        int i = idx / F;
        float d = dinv[i];
        out[idx] = h[idx] * d * d;
    }
}

// one wave per edge: lane handles a float4 feature chunk; 8 edges per 256-thread block
__global__ void __launch_bounds__(256) edge_scatter128(
    const float* __restrict__ h, const int* __restrict__ src,
    const int* __restrict__ dst, const float* __restrict__ dinv,
    float* __restrict__ out, int E)
{
    int e = blockIdx.x * 8 + (threadIdx.x >> 5);
    if (e >= E) return;
    int lane = threadIdx.x & 31;
    int s = src[e];
    int t = dst[e];
    float norm = dinv[s] * dinv[t];
    const float4 v = *((const float4*)(h + (size_t)s * NHID) + lane);
    float* __restrict__ op = out + (size_t)t * NHID + lane * 4;
    atomicAdd(op + 0, v.x * norm);
    atomicAdd(op + 1, v.y * norm);
    atomicAdd(op + 2, v.z * norm);
    atomicAdd(op + 3, v.w * norm);
}

__global__ void bias_relu(float* __restrict__ out, const float* __restrict__ b,
                          int total, int F) {
    int idx = blockIdx.x * blockDim.x + threadIdx.x;
    if (idx < total) {
        float v = out[idx] + b[idx % F];
        out[idx] = v > 0.0f ? v : 0.0f;
    }
}

// ---------------- driver ----------------

extern "C" void kernel_launch(void* const* d_in, const int* in_sizes, int n_in,
                              void* d_out, int out_size, void* d_ws, size_t ws_size,
                              hipStream_t stream) {
    const float* x   = (const float*)d_in[0];
    const int*   ei  = (const int*)d_in[1];
    const float* W1  = (const float*)d_in[2];
    const float* b1  = (const float*)d_in[3];
    const float* W2  = (const float*)d_in[4];
    const float* b2  = (const float*)d_in[5];
    const float* W3  = (const float*)d_in[6];
    const float* b3  = (const float*)d_in[7];
    const float* Wfc = (const float*)d_in[8];
    const float* bfc = (const float*)d_in[9];

    const int N = in_sizes[0] / NFEAT;   // 50000
    const int E = in_sizes[1] / 2;       // 800000
    const int* src = ei;
    const int* dst = ei + E;

    float* dinv = (float*)d_ws;                    // N floats
    float* h1   = dinv + N;                        // N * NHID
    float* h2   = h1 + (size_t)N * NHID;           // N * NHID

    const int total = N * NHID;
    const int T = 256;

    // degrees -> deg_inv_sqrt
    zero_f32      <<<(N + T - 1) / T, T, 0, stream>>>(dinv, N);
    deg_accum     <<<(E + T - 1) / T, T, 0, stream>>>(dst, dinv, E);
    deg_inv_sqrt_k<<<(N + T - 1) / T, T, 0, stream>>>(dinv, N);

    const int MB = N / (16 * MROWS);   // 625 exact (50000 = 625*80)
    dim3 gHid(MB, NHID / 16);          // 625 x 8
    dim3 gFc(MB, (NCLASS + 15) / 16);  // 625 x 1
    dim3 gScatter((E + 7) / 8);

    // ---- layer 1: x(256) -> h1, aggregate -> h2
    gemm_wmma_f32<<<gHid, 32, 0, stream>>>(x, W1, h1, NFEAT, NHID, NHID, nullptr);
    agg_init     <<<(total + T - 1) / T, T, 0, stream>>>(h1, dinv, h2, total, NHID);
    edge_scatter128<<<gScatter, T, 0, stream>>>(h1, src, dst, dinv, h2, E);
    bias_relu    <<<(total + T - 1) / T, T, 0, stream>>>(h2, b1, total, NHID);

    // ---- layer 2: h2 -> h1, aggregate -> h2
    gemm_wmma_f32<<<gHid, 32, 0, stream>>>(h2, W2, h1, NHID, NHID, NHID, nullptr);
    agg_init     <<<(total + T - 1) / T, T, 0, stream>>>(h1, dinv, h2, total, NHID);
    edge_scatter128<<<gScatter, T, 0, stream>>>(h1, src, dst, dinv, h2, E);
    bias_relu    <<<(total + T - 1) / T, T, 0, stream>>>(h2, b2, total, NHID);

    // ---- layer 3: h2 -> h1, aggregate -> h2
    gemm_wmma_f32<<<gHid, 32, 0, stream>>>(h2, W3, h1, NHID, NHID, NHID, nullptr);
    agg_init     <<<(total + T - 1) / T, T, 0, stream>>>(h1, dinv, h2, total, NHID);
    edge_scatter128<<<gScatter, T, 0, stream>>>(h1, src, dst, dinv, h2, E);
    bias_relu    <<<(total + T - 1) / T, T, 0, stream>>>(h2, b3, total, NHID);

    // ---- fc: h2(128) @ Wfc(128x8) + bfc -> d_out (masked columns)
    gemm_wmma_f32<<<gFc, 32, 0, stream>>>(h2, Wfc, (float*)d_out, NHID, NCLASS, NCLASS, bfc);
}